// L1SSIM_76673756168815
// MI455X (gfx1250) — compile-verified
//
#include <hip/hip_runtime.h>
#include <math.h>

// SSIM(11x11 Gaussian, sigma=1.5) + L1 fused loss for 64x1x512x512 fp32.
// Separable Gaussian as banded matmuls on V_WMMA_F32_16X16X4_F32 (wave32).
//
// d_ws layout (floats):
//   [0, NBLK*2)            per-block partials (ssim_sum, l1_sum)
//   [NBLK*2, NBLK*2+512)   per-lane Gaussian band fragments wf[32][16]

typedef __attribute__((ext_vector_type(2))) float v2f;
typedef __attribute__((ext_vector_type(8))) float v8f;

#define IMG     512
#define WPB     2              // waves per block
#define NTILES  65536          // 64 images * (512/16)^2
#define NBLK    (NTILES / WPB) // 32768 blocks
#define WF_OFF  (NBLK * 2)     // float offset of weight-fragment table in ws

__device__ __forceinline__ v8f wmma4(v2f a, v2f b, v8f c) {
  // D(16x16,f32) = A(16x4,f32) * B(4x16,f32) + C
  return __builtin_amdgcn_wmma_f32_16x16x4_f32(
      /*neg_a=*/false, a, /*neg_b=*/false, b,
      /*c_mod=*/(short)0, c, /*reuse_a=*/false, /*reuse_b=*/false);
}

// One wave: precompute per-lane band-matrix fragments wf[s] = g[4s + k - (lane%16)].
// These serve as the B operand (horizontal pass) AND the A operand (vertical
// pass) by the ISA's 32-bit A/B per-lane layouts.
__global__ void wf_setup_kernel(float* __restrict__ table) {
  const int lane = threadIdx.x & 31;
  const int n  = lane & 15;
  const int k0 = (lane >> 4) << 1;
  float gsum = 0.f;
  #pragma unroll
  for (int i = 0; i < 11; ++i) {
    float d = (float)i - 5.f;
    gsum += expf(-(d * d) / 4.5f);   // 2*sigma^2 = 4.5
  }
  #pragma unroll
  for (int s = 0; s < 7; ++s) {
    int j0 = 4 * s + k0 - n;
    float d0 = (float)j0 - 5.f;
    float d1 = (float)(j0 + 1) - 5.f;
    float e0 = expf(-(d0 * d0) / 4.5f) / gsum;
    float e1 = expf(-(d1 * d1) / 4.5f) / gsum;
    table[lane * 16 + 2 * s]     = (j0 >= 0 && j0 <= 10) ? e0 : 0.f;
    table[lane * 16 + 2 * s + 1] = (j0 + 1 >= 0 && j0 + 1 <= 10) ? e1 : 0.f;
  }
}

__global__ __launch_bounds__(WPB * 32)
void ssim_tile_kernel(const float* __restrict__ img1,
                      const float* __restrict__ img2,
                      const float* __restrict__ wtab,
                      float* __restrict__ partials) {
  // Per-wave staging: zero-padded input halo (26x26 valid in 32x28) for both
  // images, and the horizontally-blurred intermediate transposed: h_t[ch][n][j].
  __shared__ float s_in[WPB][2][32][28];
  __shared__ float s_ht[WPB][5][16][30];   // stride 30 (even -> 8B-aligned pairs)
  __shared__ float s_red[WPB][2];

  const int tid  = threadIdx.x;
  const int w    = tid >> 5;   // wave in block
  const int lane = tid & 31;
  const int n    = lane & 15;  // N (or M) index within fragment
  const int hi   = lane >> 4;  // high half of wave
  const int k0   = hi * 2;     // K base for this lane half (A/B 32-bit layouts)

  const int tile = blockIdx.x * WPB + w;
  const int im   = tile >> 10;        // image index
  const int t    = tile & 1023;
  const int y0   = (t >> 5) << 4;     // tile origin
  const int x0   = (t & 31) << 4;
  const long base = (long)im * (IMG * IMG);

  // --- Load precomputed Gaussian band fragments (7 aligned b64 loads) ---
  v2f wf[7];
  #pragma unroll
  for (int s = 0; s < 7; ++s)
    wf[s] = *(const v2f*)&wtab[lane * 16 + 2 * s];

  // --- Stage halo into LDS with zero padding (matches conv zero-pad) ---
  for (int r = 0; r < 32; ++r) {
    if (lane < 28) {
      int y = y0 - 5 + r;
      int x = x0 - 5 + lane;
      bool ok = (r < 26) && (lane < 26) &&
                ((unsigned)y < (unsigned)IMG) && ((unsigned)x < (unsigned)IMG);
      long idx = base + (long)y * IMG + x;
      s_in[w][0][r][lane] = ok ? img1[idx] : 0.f;
      s_in[w][1][r][lane] = ok ? img2[idx] : 0.f;
    }
  }
  // Zero H columns j=26,27 (hit by the zero-weight tail of K-slab 6; a NaN
  // there would poison every output row via 0*NaN).
  if (lane < 16) {
    v2f z; z.x = 0.f; z.y = 0.f;
    #pragma unroll
    for (int c = 0; c < 5; ++c)
      *(v2f*)&s_ht[w][c][lane][26] = z;
  }

  // --- Horizontal pass: H[r][n] = sum_c In[r][c] * g[c-n], rows 0..25 ---
  #pragma unroll
  for (int gdx = 0; gdx < 2; ++gdx) {
    const int rbase = gdx * 16;
    const int row = rbase + n;        // A-matrix row held by this lane
    v8f acc0 = {}, acc1 = {}, acc2 = {}, acc3 = {}, acc4 = {};
    #pragma unroll
    for (int s = 0; s < 7; ++s) {
      int c0 = 4 * s + k0;
      float p0 = s_in[w][0][row][c0];
      float p1 = s_in[w][0][row][c0 + 1];
      float q0 = s_in[w][1][row][c0];
      float q1 = s_in[w][1][row][c0 + 1];
      v2f a;
      a.x = p0;        a.y = p1;        acc0 = wmma4(a, wf[s], acc0); // img1
      a.x = q0;        a.y = q1;        acc1 = wmma4(a, wf[s], acc1); // img2
      a.x = p0 * p0;   a.y = p1 * p1;   acc2 = wmma4(a, wf[s], acc2); // img1^2
      a.x = q0 * q0;   a.y = q1 * q1;   acc3 = wmma4(a, wf[s], acc3); // img2^2
      a.x = p0 * q0;   a.y = p1 * q1;   acc4 = wmma4(a, wf[s], acc4); // img1*img2
    }
    // Store D to transposed H: rows (rr, rr+1) are contiguous per lane -> b64.
    #pragma unroll
    for (int v = 0; v < 8; v += 2) {
      int rr = rbase + v + 8 * hi;    // C/D layout: vgpr v, lane half -> row
      if (rr < 25) {                  // keep rows 0..25 only
        v2f p;
        p.x = acc0[v]; p.y = acc0[v + 1]; *(v2f*)&s_ht[w][0][n][rr] = p;
        p.x = acc1[v]; p.y = acc1[v + 1]; *(v2f*)&s_ht[w][1][n][rr] = p;
        p.x = acc2[v]; p.y = acc2[v + 1]; *(v2f*)&s_ht[w][2][n][rr] = p;
        p.x = acc3[v]; p.y = acc3[v + 1]; *(v2f*)&s_ht[w][3][n][rr] = p;
        p.x = acc4[v]; p.y = acc4[v + 1]; *(v2f*)&s_ht[w][4][n][rr] = p;
      }
    }
  }

  // --- Vertical pass: Out[m][n] = sum_j g[j-m] * H[j][n] ---
  v8f m1 = {}, m2 = {}, x11 = {}, x22 = {}, x12 = {};
  #pragma unroll
  for (int s = 0; s < 7; ++s) {
    int j0 = 4 * s + k0;              // B layout: this lane half holds K=j0,j0+1
    v2f b;
    b = *(const v2f*)&s_ht[w][0][n][j0]; m1  = wmma4(wf[s], b, m1);
    b = *(const v2f*)&s_ht[w][1][n][j0]; m2  = wmma4(wf[s], b, m2);
    b = *(const v2f*)&s_ht[w][2][n][j0]; x11 = wmma4(wf[s], b, x11);
    b = *(const v2f*)&s_ht[w][3][n][j0]; x22 = wmma4(wf[s], b, x22);
    b = *(const v2f*)&s_ht[w][4][n][j0]; x12 = wmma4(wf[s], b, x12);
  }

  // --- SSIM map + L1 over this tile ---
  const float C1 = 1e-4f, C2 = 9e-4f;
  float ssim_sum = 0.f, l1_sum = 0.f;
  #pragma unroll
  for (int v = 0; v < 8; ++v) {
    float mu1 = m1[v], mu2 = m2[v];
    float mu1sq = mu1 * mu1, mu2sq = mu2 * mu2, mu12 = mu1 * mu2;
    float sg1 = x11[v] - mu1sq;
    float sg2 = x22[v] - mu2sq;
    float sg12 = x12[v] - mu12;
    float num = (2.f * mu12 + C1) * (2.f * sg12 + C2);
    float den = (mu1sq + mu2sq + C1) * (sg1 + sg2 + C2);
    ssim_sum += num / den;
    int m = v + 8 * hi;               // pixel (y0+m, x0+n) = halo (5+m, 5+n)
    float p = s_in[w][0][5 + m][5 + n];
    float q = s_in[w][1][5 + m][5 + n];
    l1_sum += fabsf(p - q);
  }

  // --- Deterministic reduction: wave shfl_xor, then per-block partials ---
  #pragma unroll
  for (int off = 16; off > 0; off >>= 1) {
    ssim_sum += __shfl_xor(ssim_sum, off, 32);
    l1_sum   += __shfl_xor(l1_sum, off, 32);
  }
  if (lane == 0) { s_red[w][0] = ssim_sum; s_red[w][1] = l1_sum; }
  __syncthreads();
  if (tid == 0) {
    float ps = 0.f, pl = 0.f;
    #pragma unroll
    for (int i = 0; i < WPB; ++i) { ps += s_red[i][0]; pl += s_red[i][1]; }
    partials[2 * blockIdx.x]     = ps;
    partials[2 * blockIdx.x + 1] = pl;
  }
}

__global__ void final_reduce(const float* __restrict__ partials,
                             float* __restrict__ out) {
  __shared__ double sh[256][2];
  double a = 0.0, b = 0.0;
  for (int i = threadIdx.x; i < NBLK; i += 256) {
    a += (double)partials[2 * i];
    b += (double)partials[2 * i + 1];
  }
  sh[threadIdx.x][0] = a;
  sh[threadIdx.x][1] = b;
  __syncthreads();
  for (int off = 128; off > 0; off >>= 1) {
    if ((int)threadIdx.x < off) {
      sh[threadIdx.x][0] += sh[threadIdx.x + off][0];
      sh[threadIdx.x][1] += sh[threadIdx.x + off][1];
    }
    __syncthreads();
  }
  if (threadIdx.x == 0) {
    const double inv_n = 1.0 / 16777216.0;  // 64*1*512*512
    double ssim = sh[0][0] * inv_n;
    double l1   = sh[0][1] * inv_n;
    out[0] = (float)(0.5 * (1.0 - ssim) + 0.5 * l1);
  }
}

extern "C" void kernel_launch(void* const* d_in, const int* in_sizes, int n_in,
                              void* d_out, int out_size, void* d_ws, size_t ws_size,
                              hipStream_t stream) {
  (void)in_sizes; (void)n_in; (void)out_size; (void)ws_size;
  const float* img1 = (const float*)d_in[0];
  const float* img2 = (const float*)d_in[1];
  float* ws         = (float*)d_ws;        // needs (NBLK*2 + 512) floats
  float* partials   = ws;
  float* wtab       = ws + WF_OFF;
  float* out        = (float*)d_out;

  wf_setup_kernel<<<dim3(1), dim3(32), 0, stream>>>(wtab);
  ssim_tile_kernel<<<dim3(NBLK), dim3(WPB * 32), 0, stream>>>(img1, img2, wtab, partials);
  final_reduce<<<dim3(1), dim3(256), 0, stream>>>(partials, out);
}